// DyWAN_80985903333560
// MI455X (gfx1250) — compile-verified
//
#include <hip/hip_runtime.h>
#include <math.h>

typedef __attribute__((ext_vector_type(2))) float v2f;
typedef __attribute__((ext_vector_type(4))) float v4f;
typedef __attribute__((ext_vector_type(8))) float v8f;

#define FILTER_LEN 8
#define HIDDEN 32
#define CHANNEL 512
#define BATCH 256
#define SEQLEN 2048

// ---------------------------------------------------------------------------
// Kernel 1: stat[b*C+c] = max over L of x[b][c][:].  1 GiB streamed once.
// One wave32 per row of 2048 floats: 16 independent b128 NT loads per lane,
// then a 5-step shuffle max-reduction.
// ---------------------------------------------------------------------------
__global__ void __launch_bounds__(256)
dywan_rowmax_kernel(const float* __restrict__ x, float* __restrict__ stat) {
    const int lane = threadIdx.x & 31;
    const int wid  = threadIdx.x >> 5;                    // 8 waves / block
    const size_t row = (size_t)blockIdx.x * 8 + wid;      // 131072 rows total
    const v4f* p = (const v4f*)x + row * (SEQLEN / 4);

    float m = -INFINITY;
#pragma unroll
    for (int i = 0; i < 16; ++i) {
        v4f v = __builtin_nontemporal_load(&p[i * 32 + lane]);
        m = fmaxf(m, fmaxf(fmaxf(v.x, v.y), fmaxf(v.z, v.w)));
    }
#pragma unroll
    for (int off = 16; off >= 1; off >>= 1)
        m = fmaxf(m, __shfl_xor(m, off, 32));
    if (lane == 0) stat[row] = m;
}

// ---------------------------------------------------------------------------
// Kernel 2: 3-layer MLP via V_WMMA_F32_16X16X4_F32, one wave per 16 samples.
//   h1 = relu(stat @ W1^T + b1)   (256x512)x(512x32)
//   h2 = relu(h1   @ W2^T + b2)   (256x32) x(32x32)
//   f  =      h2   @ W3^T + b3    (256x32) x(32x16)
// lo = f[:, :8] -> out[0..2047], hi = f[:, 8:] -> out[2048..4095]
// ---------------------------------------------------------------------------
static __device__ __forceinline__ v8f wmma_f32_4(v2f a, v2f b, v8f c) {
    return __builtin_amdgcn_wmma_f32_16x16x4_f32(
        /*neg_a=*/false, a, /*neg_b=*/false, b,
        /*c_mod=*/(short)0, c, /*reuse_a=*/false, /*reuse_b=*/false);
}

__global__ void __launch_bounds__(32)
dywan_mlp_wmma_kernel(const float* __restrict__ stat,
                      const float* __restrict__ W1, const float* __restrict__ b1,
                      const float* __restrict__ W2, const float* __restrict__ b2,
                      const float* __restrict__ W3, const float* __restrict__ b3,
                      float* __restrict__ out) {
    __shared__ float hbuf[16][HIDDEN + 1];   // padded to dodge bank conflicts

    const int lane = threadIdx.x;            // 0..31, full wave (EXEC all 1s)
    const int half = lane >> 4;              // 0: K={0,1}, 1: K={2,3} (A/B layout)
    const int l16  = lane & 15;
    const int m0   = blockIdx.x * 16;        // 16 samples per wave

    // ---------------- GEMM1: K=512, two N-tiles of 16 ----------------
    v8f c0 = {}; v8f c1 = {};
    const float* arow = stat + (size_t)(m0 + l16) * CHANNEL + half * 2;  // A: M=l16
    const float* brw0 = W1 + (size_t)(l16)      * CHANNEL + half * 2;    // B tile n0=0
    const float* brw1 = W1 + (size_t)(16 + l16) * CHANNEL + half * 2;    // B tile n0=16
#pragma unroll 8
    for (int kk = 0; kk < CHANNEL; kk += 4) {
        v2f a  = *(const v2f*)(arow + kk);
        v2f bA = *(const v2f*)(brw0 + kk);
        v2f bB = *(const v2f*)(brw1 + kk);
        c0 = wmma_f32_4(a, bA, c0);
        c1 = wmma_f32_4(a, bB, c1);
    }
    // bias + relu, scatter C layout (row = v + 8*half, col = l16) into LDS
    {
        float bia0 = b1[l16], bia1 = b1[16 + l16];
#pragma unroll
        for (int v = 0; v < 8; ++v) {
            int row = v + half * 8;
            hbuf[row][l16]      = fmaxf(c0[v] + bia0, 0.0f);
            hbuf[row][16 + l16] = fmaxf(c1[v] + bia1, 0.0f);
        }
    }
    __syncthreads();

    // ---------------- GEMM2: K=32, two N-tiles ----------------
    v8f d0 = {}; v8f d1 = {};
#pragma unroll
    for (int kk = 0; kk < HIDDEN; kk += 4) {
        v2f a;
        a.x = hbuf[l16][kk + half * 2];
        a.y = hbuf[l16][kk + half * 2 + 1];
        v2f bA = *(const v2f*)(W2 + (size_t)(l16)      * HIDDEN + kk + half * 2);
        v2f bB = *(const v2f*)(W2 + (size_t)(16 + l16) * HIDDEN + kk + half * 2);
        d0 = wmma_f32_4(a, bA, d0);
        d1 = wmma_f32_4(a, bB, d1);
    }
    __syncthreads();   // done reading hbuf before overwrite
    {
        float bia0 = b2[l16], bia1 = b2[16 + l16];
#pragma unroll
        for (int v = 0; v < 8; ++v) {
            int row = v + half * 8;
            hbuf[row][l16]      = fmaxf(d0[v] + bia0, 0.0f);
            hbuf[row][16 + l16] = fmaxf(d1[v] + bia1, 0.0f);
        }
    }
    __syncthreads();

    // ---------------- GEMM3: K=32, one N-tile of 16 ----------------
    v8f f = {};
#pragma unroll
    for (int kk = 0; kk < HIDDEN; kk += 4) {
        v2f a;
        a.x = hbuf[l16][kk + half * 2];
        a.y = hbuf[l16][kk + half * 2 + 1];
        v2f b = *(const v2f*)(W3 + (size_t)(l16) * HIDDEN + kk + half * 2);
        f = wmma_f32_4(a, b, f);
    }
    {
        float bf = b3[l16];
#pragma unroll
        for (int v = 0; v < 8; ++v) {
            int m = m0 + v + half * 8;
            int n = l16;
            float val = f[v] + bf;
            if (n < FILTER_LEN)
                out[(size_t)m * FILTER_LEN + n] = val;                            // lo_f
            else
                out[BATCH * FILTER_LEN + (size_t)m * FILTER_LEN + (n - 8)] = val; // hi_f
        }
    }
}

// ---------------------------------------------------------------------------
// Kernel 3: ortho loss. One thread per sample, block-reduce, single writer.
// ---------------------------------------------------------------------------
__global__ void __launch_bounds__(256)
dywan_ortho_loss_kernel(const float* __restrict__ lohi, float* __restrict__ loss) {
    __shared__ float red[256];
    const int m = threadIdx.x;

    float L[FILTER_LEN], H[FILTER_LEN];
#pragma unroll
    for (int i = 0; i < FILTER_LEN; ++i) {
        L[i] = lohi[(size_t)m * FILTER_LEN + i];
        H[i] = lohi[BATCH * FILTER_LEN + (size_t)m * FILTER_LEN + i];
    }
    float nl = 0.0f, nh = 0.0f;
#pragma unroll
    for (int i = 0; i < FILTER_LEN; ++i) { nl += L[i] * L[i]; nh += H[i] * H[i]; }
    float il = 1.0f / sqrtf(nl), ih = 1.0f / sqrtf(nh);
#pragma unroll
    for (int i = 0; i < FILTER_LEN; ++i) { L[i] *= il; H[i] *= ih; }

    float s = 0.0f;
#pragma unroll
    for (int sh = 1; sh < FILTER_LEN; sh += 2) {      // shifts 1,3,5,7
        float d = 0.0f;
#pragma unroll
        for (int i = 0; i < FILTER_LEN; ++i)
            d += L[i] * L[(i - sh) & (FILTER_LEN - 1)];   // roll(Ln, sh)
        s += fabsf(d);
    }
    float dlh = 0.0f, dll = 0.0f, dhh = 0.0f;
#pragma unroll
    for (int i = 0; i < FILTER_LEN; ++i) {
        dlh += L[i] * H[i]; dll += L[i] * L[i]; dhh += H[i] * H[i];
    }
    s += fabsf(dlh) + fabsf(dll - 1.0f) + fabsf(dhh - 1.0f);

    red[m] = s;
    __syncthreads();
#pragma unroll
    for (int off = 128; off >= 1; off >>= 1) {
        if (m < off) red[m] += red[m + off];
        __syncthreads();
    }
    if (m == 0) loss[0] = red[0] / (float)BATCH;
}

// ---------------------------------------------------------------------------
extern "C" void kernel_launch(void* const* d_in, const int* in_sizes, int n_in,
                              void* d_out, int out_size, void* d_ws, size_t ws_size,
                              hipStream_t stream) {
    const float* x  = (const float*)d_in[0];
    const float* W1 = (const float*)d_in[1];
    const float* b1 = (const float*)d_in[2];
    const float* W2 = (const float*)d_in[3];
    const float* b2 = (const float*)d_in[4];
    const float* W3 = (const float*)d_in[5];
    const float* b3 = (const float*)d_in[6];
    float* out  = (float*)d_out;
    float* stat = (float*)d_ws;     // 256*512 floats = 512 KB scratch

    // 131072 rows, 8 waves (rows) per 256-thread block -> 16384 blocks
    dywan_rowmax_kernel<<<(BATCH * CHANNEL) / 8, 256, 0, stream>>>(x, stat);
    // 256 samples / 16 per wave -> 16 single-wave blocks
    dywan_mlp_wmma_kernel<<<BATCH / 16, 32, 0, stream>>>(stat, W1, b1, W2, b2,
                                                         W3, b3, out);
    dywan_ortho_loss_kernel<<<1, 256, 0, stream>>>(out, out + 2 * BATCH * FILTER_LEN);
}